// QLSTMHybrid_65481071397229
// MI455X (gfx1250) — compile-verified
//
#include <hip/hip_runtime.h>
#include <hip/hip_bf16.h>
#include <math.h>

typedef __attribute__((ext_vector_type(2))) float v2f;
typedef __attribute__((ext_vector_type(8))) float v8f;

#define NQb   8
#define NCOL  32      // 4 gates * 8 wires
#define E_DIM 1024
#define D_DIM 1032    // E + H

// ---------------------------------------------------------------------------
// Phase 1: pre[t][col] = emb[sentence[t]] . W_col[0:1024] + b_col
// col: 0..7 = f, 8..15 = i, 16..23 = u, 24..31 = o  (wire = col & 7)
// One wave per 16-row tile; fp32 WMMA 16x16x4, two N-tiles of 16 columns.
// A 16x4 layout : lane(lo)=M, hi half -> K+2..3 ; per-lane {A[m][k],A[m][k+1]}
// B 4x16 layout : lane(lo)=N, hi half -> K+2..3 ; per-lane {B[k][n],B[k+1][n]}
// D 16x16 layout: acc[r] -> row r + 8*hi, col lo
// ---------------------------------------------------------------------------
#if __has_builtin(__builtin_amdgcn_wmma_f32_16x16x4_f32)
#define QLSTM_HAVE_WMMA_F32 1
__global__ void qlstm_gemm_wmma(const int* __restrict__ sentence,
                                const float* __restrict__ emb,
                                const float* __restrict__ Wf, const float* __restrict__ Wi,
                                const float* __restrict__ Wu, const float* __restrict__ Wo,
                                const float* __restrict__ bf, const float* __restrict__ bi,
                                const float* __restrict__ bu, const float* __restrict__ bo,
                                float* __restrict__ pre, int T)
{
    const int lane = threadIdx.x;       // 32 threads = 1 wave
    const int hi   = lane >> 4;         // 0/1 -> K offset 0/2
    const int lo   = lane & 15;
    const int tile = blockIdx.x;

    int trow = tile * 16 + lo;
    if (trow >= T) trow = T - 1;        // clamp; excess stores are guarded below
    const float* arow = emb + (size_t)sentence[trow] * E_DIM;

    const int w0 = lo & 7;
    const float* W0 = ((lo < 8) ? Wf : Wi) + (size_t)w0 * D_DIM;   // cols  0..15
    const float* W1 = ((lo < 8) ? Wu : Wo) + (size_t)w0 * D_DIM;   // cols 16..31

    v8f acc0 = {};
    v8f acc1 = {};

    for (int k0 = 0; k0 < E_DIM; k0 += 4) {
        const int k = k0 + 2 * hi;
        v2f a  = *(const v2f*)(arow + k);
        v2f b0 = *(const v2f*)(W0 + k);
        v2f b1 = *(const v2f*)(W1 + k);
        acc0 = __builtin_amdgcn_wmma_f32_16x16x4_f32(false, a, false, b0,
                                                     (short)0, acc0, false, false);
        acc1 = __builtin_amdgcn_wmma_f32_16x16x4_f32(false, a, false, b1,
                                                     (short)0, acc1, false, false);
    }

    const float bias0 = (lo < 8) ? bf[w0] : bi[w0];
    const float bias1 = (lo < 8) ? bu[w0] : bo[w0];

#pragma unroll
    for (int r = 0; r < 8; ++r) {
        const int row = tile * 16 + r + 8 * hi;
        if (row < T) {
            pre[(size_t)row * NCOL + lo]      = acc0[r] + bias0;
            pre[(size_t)row * NCOL + 16 + lo] = acc1[r] + bias1;
        }
    }
}
#else
// Fallback: plain FMA GEMV per output element (only if f32 WMMA builtin absent)
__global__ void qlstm_gemm_fallback(const int* __restrict__ sentence,
                                    const float* __restrict__ emb,
                                    const float* __restrict__ Wf, const float* __restrict__ Wi,
                                    const float* __restrict__ Wu, const float* __restrict__ Wo,
                                    const float* __restrict__ bf, const float* __restrict__ bi,
                                    const float* __restrict__ bu, const float* __restrict__ bo,
                                    float* __restrict__ pre, int T)
{
    int idx = blockIdx.x * blockDim.x + threadIdx.x;
    if (idx >= T * NCOL) return;
    int t = idx / NCOL, col = idx % NCOL, w = col & 7, g = col >> 3;
    const float* W = (g == 0) ? Wf : (g == 1) ? Wi : (g == 2) ? Wu : Wo;
    const float* b = (g == 0) ? bf : (g == 1) ? bi : (g == 2) ? bu : bo;
    const float* x = emb + (size_t)sentence[t] * E_DIM;
    const float* wr = W + (size_t)w * D_DIM;
    float a = b[w];
    for (int k = 0; k < E_DIM; ++k) a = fmaf(x[k], wr[k], a);
    pre[(size_t)t * NCOL + col] = a;
}
#endif

// ---------------------------------------------------------------------------
// Phase 2: sequential LSTM scan. One wave; lane = gate*8 + wire.
// Circuit collapses to z[w] = prod_{u<=w} cos(theta_u)*cos(rx_u)
// (product-state + GF(2)-linear CNOT chain; see analysis).
// Cross-lane traffic via __shfl (ds_bpermute), zero barriers.
// ---------------------------------------------------------------------------
__global__ void qlstm_scan(const float* __restrict__ pre,
                           const float* __restrict__ Wf, const float* __restrict__ Wi,
                           const float* __restrict__ Wu, const float* __restrict__ Wo,
                           const float* __restrict__ rxf, const float* __restrict__ rxi,
                           const float* __restrict__ rxu, const float* __restrict__ rxo,
                           float* __restrict__ outs, int T)
{
    const int lane = threadIdx.x;       // 0..31
    const int g = lane >> 3;            // gate: 0=f 1=i 2=u 3=o
    const int w = lane & 7;             // wire

    const float* Wg  = (g == 0) ? Wf  : (g == 1) ? Wi  : (g == 2) ? Wu  : Wo;
    const float* rxg = (g == 0) ? rxf : (g == 1) ? rxi : (g == 2) ? rxu : rxo;

    float whh[8];
#pragma unroll
    for (int j = 0; j < 8; ++j) whh[j] = Wg[(size_t)w * D_DIM + E_DIM + j];
    const float rc = cosf(rxg[w]);

    float h[8];
#pragma unroll
    for (int j = 0; j < 8; ++j) h[j] = 0.f;
    float cc = 0.f;

    for (int t = 0; t < T; ++t) {
        // theta = pre-computed x-part + h-part
        float a = pre[(size_t)t * NCOL + lane];
#pragma unroll
        for (int j = 0; j < 8; ++j) a = fmaf(h[j], whh[j], a);

        // per-wire factor, then inclusive product-scan within 8-lane group
        float cum = cosf(a) * rc;
#pragma unroll
        for (int d = 1; d < 8; d <<= 1) {
            float o = __shfl(cum, lane - d, 32);
            if (w >= d) cum *= o;
        }

        // activation: sigmoid for f,i,o ; tanh for u
        float act = (g == 2) ? tanhf(cum) : 1.f / (1.f + expf(-cum));

        float fv = __shfl(act,      w, 32);
        float iv = __shfl(act,  8 + w, 32);
        float gv = __shfl(act, 16 + w, 32);
        float ov = __shfl(act, 24 + w, 32);

        cc = fmaf(fv, cc, iv * gv);          // replicated per-wire across gates
        float hv = ov * tanhf(cc);

        if (g == 0) outs[(size_t)t * NQb + w] = hv;
#pragma unroll
        for (int j = 0; j < 8; ++j) h[j] = __shfl(hv, j, 32);
    }
}

// ---------------------------------------------------------------------------
// Phase 3: tag head — parallel over T.
// z[w] = prod_{u<=w} cos(tag_logits[u]);  out = log((z+1)/2 + 1e-12)
// ---------------------------------------------------------------------------
__global__ void qlstm_tag(const float* __restrict__ outs,
                          const float* __restrict__ Wtag,
                          const float* __restrict__ btag,
                          float* __restrict__ out, int T)
{
    int t = blockIdx.x * blockDim.x + threadIdx.x;
    if (t >= T) return;
    float x[8];
#pragma unroll
    for (int j = 0; j < 8; ++j) x[j] = outs[(size_t)t * NQb + j];
    float cum = 1.f;
#pragma unroll
    for (int w = 0; w < 8; ++w) {
        float a = btag[w];
#pragma unroll
        for (int j = 0; j < 8; ++j) a = fmaf(x[j], Wtag[w * 8 + j], a);
        cum *= cosf(a);
        out[(size_t)t * NQb + w] = logf(fmaf(cum, 0.5f, 0.5f) + 1e-12f);
    }
}

// ---------------------------------------------------------------------------
extern "C" void kernel_launch(void* const* d_in, const int* in_sizes, int n_in,
                              void* d_out, int out_size, void* d_ws, size_t ws_size,
                              hipStream_t stream)
{
    const int*   sentence = (const int*)  d_in[0];
    const float* emb      = (const float*)d_in[1];
    const float* Wf       = (const float*)d_in[2];
    const float* bf       = (const float*)d_in[3];
    const float* Wi       = (const float*)d_in[4];
    const float* bi       = (const float*)d_in[5];
    const float* Wu       = (const float*)d_in[6];
    const float* bu       = (const float*)d_in[7];
    const float* Wo       = (const float*)d_in[8];
    const float* bo       = (const float*)d_in[9];
    const float* rxf      = (const float*)d_in[10];
    const float* rxi      = (const float*)d_in[11];
    const float* rxu      = (const float*)d_in[12];
    const float* rxo      = (const float*)d_in[13];
    const float* Wtag     = (const float*)d_in[14];
    const float* btag     = (const float*)d_in[15];

    const int T = in_sizes[0];           // T*B with B==1

    float* pre  = (float*)d_ws;          // T*32 floats
    float* outs = pre + (size_t)T * NCOL;// T*8 floats

#if defined(QLSTM_HAVE_WMMA_F32)
    qlstm_gemm_wmma<<<(T + 15) / 16, 32, 0, stream>>>(
        sentence, emb, Wf, Wi, Wu, Wo, bf, bi, bu, bo, pre, T);
#else
    qlstm_gemm_fallback<<<(T * NCOL + 255) / 256, 256, 0, stream>>>(
        sentence, emb, Wf, Wi, Wu, Wo, bf, bi, bu, bo, pre, T);
#endif

    qlstm_scan<<<1, 32, 0, stream>>>(pre, Wf, Wi, Wu, Wo,
                                     rxf, rxi, rxu, rxo, outs, T);

    qlstm_tag<<<(T + 255) / 256, 256, 0, stream>>>(outs, Wtag, btag,
                                                   (float*)d_out, T);
}